// RBF_GAT_78151224918200
// MI455X (gfx1250) — compile-verified
//
#include <hip/hip_runtime.h>
#include <hip/hip_bf16.h>
#include <math.h>

// ---------------------------------------------------------------------------
// GAT layer for MI455X (gfx1250, wave32, WMMA).
// GEMMs use V_WMMA_F32_16X16X4_F32 (f32 in, f32 accum -> matches reference).
// Memory-bound problem: efeats stream (410MB) dominates; node-side buffers
// (12.8MB each) live in the 192MB L2, so gathers in the edge kernel hit L2.
// ---------------------------------------------------------------------------

typedef float v2f __attribute__((ext_vector_type(2)));
typedef float v8f __attribute__((ext_vector_type(8)));

__device__ __forceinline__ v8f wmma_f32_16x16x4(v2f a, v2f b, v8f c) {
  // 8 args: (neg_a, A, neg_b, B, c_mod, C, reuse_a, reuse_b)
  return __builtin_amdgcn_wmma_f32_16x16x4_f32(false, a, false, b, (short)0, c,
                                               false, false);
}

__device__ __forceinline__ void atomicMaxF(float* addr, float val) {
  // Sign-trick float max: works for any mix of signs when buffer starts at -inf.
  if (val >= 0.0f)
    atomicMax((int*)addr, __float_as_int(val));
  else
    atomicMin((unsigned int*)addr, __float_as_uint(val));
}

// ---------------------------------------------------------------------------
// Kernel 0: init m = -inf, s = 0, h_out = 0
// ---------------------------------------------------------------------------
__global__ void k_init(float* __restrict__ h_out, float* __restrict__ mbuf,
                       float* __restrict__ sbuf, int N) {
  int idx = blockIdx.x * blockDim.x + threadIdx.x;
  if (idx < N * 64) h_out[idx] = 0.0f;
  if (idx < N * 8) {
    mbuf[idx] = -INFINITY;
    sbuf[idx] = 0.0f;
  }
}

// ---------------------------------------------------------------------------
// Kernel 1: node projections  f_ni = X@W_ni, f_nj = X@W_nj, h = X@W_node + b
// One wave handles a 16-node x 16-col tile of one of the 3 projections.
// blockDim = 128 (4 waves). grid = (ceil(tiles/4), 12).
// ---------------------------------------------------------------------------
__global__ __launch_bounds__(128) void k_node_proj(
    const float* __restrict__ nfeats, const float* __restrict__ W_ni,
    const float* __restrict__ W_nj, const float* __restrict__ W_node,
    const float* __restrict__ b_node, float* __restrict__ f_ni,
    float* __restrict__ f_nj, float* __restrict__ hbuf, int ntiles) {
  int wv = threadIdx.x >> 5;
  int lane = threadIdx.x & 31;
  int tile = blockIdx.x * 4 + wv;            // 16 nodes per tile
  if (tile >= ntiles) return;                // wave-uniform: EXEC stays all-1s
  int task = blockIdx.y;                     // 0..11
  int mat = task >> 2;                       // 0:f_ni 1:f_nj 2:h
  int ctile = task & 3;                      // 16-column group of 64

  const float* __restrict__ W = (mat == 0) ? W_ni : (mat == 1 ? W_nj : W_node);
  float* __restrict__ outp = (mat == 0) ? f_ni : (mat == 1 ? f_nj : hbuf);

  int r = lane & 15;                         // row (A) / col (B) within tile
  int kb = (lane >> 4) << 1;                 // K-pair select (0 or 2)
  const float* __restrict__ arow = nfeats + (size_t)(tile * 16 + r) * 128;
  const float* __restrict__ bcol = W + (ctile * 16 + r);

  v8f c = {};
#pragma unroll 4
  for (int k0 = 0; k0 < 128; k0 += 4) {
    v2f a, b;
    a.x = arow[k0 + kb];
    a.y = arow[k0 + kb + 1];
    b.x = bcol[(size_t)(k0 + kb) * 64];
    b.y = bcol[(size_t)(k0 + kb + 1) * 64];
    c = wmma_f32_16x16x4(a, b, c);
  }

  // D layout: VGPR i -> row (i + 8*(lane>=16)), col = lane&15
  int outcol = ctile * 16 + (lane & 15);
  int rbase = tile * 16 + ((lane >> 4) << 3);
  float bb = (mat == 2) ? b_node[outcol] : 0.0f;
#pragma unroll
  for (int i = 0; i < 8; ++i) {
    outp[(size_t)(rbase + i) * 64 + outcol] = c[i] + bb;
  }
}

// ---------------------------------------------------------------------------
// Kernel 2: edge projection + assemble + leaky_relu + attention logits + max
// f_fij = efeats@W_fij via WMMA; f_tmp = f_fij + f_ni[src] + f_nj[dst] + bias;
// f_out = leaky_relu(f_tmp);  e[E,H] = sum_fe f_out*attn;  m = seg_max(e,dst)
// One wave per 16-edge x 16-col tile. grid = (E/16/4, 4), blockDim = 128.
// ---------------------------------------------------------------------------
__global__ __launch_bounds__(128) void k_edge_proj(
    const float* __restrict__ efeats, const float* __restrict__ W_fij,
    const float* __restrict__ f_ni, const float* __restrict__ f_nj,
    const float* __restrict__ bias, const float* __restrict__ attn,
    const int* __restrict__ src, const int* __restrict__ dst,
    float* __restrict__ f_out, float* __restrict__ ebuf,
    float* __restrict__ mbuf, int etiles) {
  int wv = threadIdx.x >> 5;
  int lane = threadIdx.x & 31;
  int tile = blockIdx.x * 4 + wv;
  if (tile >= etiles) return;
  int ctile = blockIdx.y;                    // 0..3

  int r = lane & 15;
  int kb = (lane >> 4) << 1;
  const float* __restrict__ arow = efeats + (size_t)(tile * 16 + r) * 128;
  const float* __restrict__ bcol = W_fij + (ctile * 16 + r);

  // Prefetch the next tile's A rows into cache (streaming access pattern).
  __builtin_prefetch(arow + 16 * 128, 0, 1);

  v8f c = {};
#pragma unroll 4
  for (int k0 = 0; k0 < 128; k0 += 4) {
    v2f a, b;
    a.x = arow[k0 + kb];
    a.y = arow[k0 + kb + 1];
    b.x = bcol[(size_t)(k0 + kb) * 64];
    b.y = bcol[(size_t)(k0 + kb + 1) * 64];
    c = wmma_f32_16x16x4(a, b, c);
  }

  int col64 = ctile * 16 + (lane & 15);      // column in [0,64)
  int hh = col64 >> 3;                       // head
  int fe = col64 & 7;                        // feature within head
  float av = attn[hh * 8 + fe];              // attn is [1,8,8] flat
  float bv = bias[col64];
  int rbase = tile * 16 + ((lane >> 4) << 3);

#pragma unroll
  for (int i = 0; i < 8; ++i) {
    int edge = rbase + i;
    int sN = src[edge];
    int dN = dst[edge];
    float v = c[i] + f_ni[(size_t)sN * 64 + col64] +
              f_nj[(size_t)dN * 64 + col64] + bv;
    v = (v > 0.0f) ? v : 0.01f * v;          // leaky_relu (slope 0.01)
    __builtin_nontemporal_store(v, &f_out[(size_t)edge * 64 + col64]);
    // reduce over fe (8 aligned lanes) for this head's logit
    float t = v * av;
    t += __shfl_xor(t, 1, 32);
    t += __shfl_xor(t, 2, 32);
    t += __shfl_xor(t, 4, 32);
    if (fe == 0) {
      ebuf[(size_t)edge * 8 + hh] = t;
      atomicMaxF(&mbuf[(size_t)dN * 8 + hh], t);
    }
  }
}

// ---------------------------------------------------------------------------
// Kernel 3: ex = exp(e - m[dst]) (in place) ; s = segment_sum(ex, dst)
// ---------------------------------------------------------------------------
__global__ void k_softmax_num(float* __restrict__ ebuf,
                              const float* __restrict__ mbuf,
                              float* __restrict__ sbuf,
                              const int* __restrict__ dst, int E) {
  int idx = blockIdx.x * blockDim.x + threadIdx.x;
  if (idx >= E * 8) return;
  int edge = idx >> 3;
  int hh = idx & 7;
  int dN = dst[edge];
  float ex = __expf(ebuf[idx] - mbuf[dN * 8 + hh]);
  ebuf[idx] = ex;
  atomicAdd(&sbuf[dN * 8 + hh], ex);
}

// ---------------------------------------------------------------------------
// Kernel 4: a = ex / s[dst] ; h_out += h[src] * a  (scatter to dst)
// ---------------------------------------------------------------------------
__global__ void k_scatter(const float* __restrict__ ebuf,
                          const float* __restrict__ sbuf,
                          const float* __restrict__ hbuf,
                          const int* __restrict__ src,
                          const int* __restrict__ dst,
                          float* __restrict__ h_out, int E) {
  long long idx = (long long)blockIdx.x * blockDim.x + threadIdx.x;
  if (idx >= (long long)E * 64) return;
  int edge = (int)(idx >> 6);
  int cc = (int)(idx & 63);
  int hh = cc >> 3;
  int sN = src[edge];
  int dN = dst[edge];
  float a = ebuf[(size_t)edge * 8 + hh] / sbuf[(size_t)dN * 8 + hh];
  atomicAdd(&h_out[(size_t)dN * 64 + cc], hbuf[(size_t)sN * 64 + cc] * a);
}

// ---------------------------------------------------------------------------
// Launcher
// Inputs: 0 nfeats[N,128] 1 efeats[E,128] 2 src[E] 3 dst[E] 4 W_node[128,64]
//         5 b_node[64] 6 W_ni[128,64] 7 W_nj[128,64] 8 W_fij[128,64]
//         9 attn[1,8,8] 10 bias[64]
// Output: concat(h_out[N,8,8], f_out[E,8,8]) flat f32
// ---------------------------------------------------------------------------
extern "C" void kernel_launch(void* const* d_in, const int* in_sizes, int n_in,
                              void* d_out, int out_size, void* d_ws,
                              size_t ws_size, hipStream_t stream) {
  const float* nfeats = (const float*)d_in[0];
  const float* efeats = (const float*)d_in[1];
  const int* src = (const int*)d_in[2];
  const int* dst = (const int*)d_in[3];
  const float* W_node = (const float*)d_in[4];
  const float* b_node = (const float*)d_in[5];
  const float* W_ni = (const float*)d_in[6];
  const float* W_nj = (const float*)d_in[7];
  const float* W_fij = (const float*)d_in[8];
  const float* attn = (const float*)d_in[9];
  const float* bias = (const float*)d_in[10];

  int N = in_sizes[0] / 128;
  int E = in_sizes[1] / 128;

  float* out = (float*)d_out;
  float* h_out = out;                         // [N,64]
  float* f_out = out + (size_t)N * 64;        // [E,64]

  // Workspace layout (floats)
  float* ws = (float*)d_ws;
  float* f_ni = ws;                           // N*64
  float* f_nj = f_ni + (size_t)N * 64;        // N*64
  float* hbuf = f_nj + (size_t)N * 64;        // N*64
  float* ebuf = hbuf + (size_t)N * 64;        // E*8 (logits, then ex in place)
  float* mbuf = ebuf + (size_t)E * 8;         // N*8
  float* sbuf = mbuf + (size_t)N * 8;         // N*8

  // 0) init
  {
    int total = N * 64;
    k_init<<<(total + 255) / 256, 256, 0, stream>>>(h_out, mbuf, sbuf, N);
  }
  // 1) node projections (WMMA)
  {
    int ntiles = (N + 15) / 16;
    dim3 grid((ntiles + 3) / 4, 12);
    k_node_proj<<<grid, 128, 0, stream>>>(nfeats, W_ni, W_nj, W_node, b_node,
                                          f_ni, f_nj, hbuf, ntiles);
  }
  // 2) edge projection + logits + segment max (WMMA)
  {
    int etiles = (E + 15) / 16;
    dim3 grid((etiles + 3) / 4, 4);
    k_edge_proj<<<grid, 128, 0, stream>>>(efeats, W_fij, f_ni, f_nj, bias, attn,
                                          src, dst, f_out, ebuf, mbuf, etiles);
  }
  // 3) softmax numerator + segment sum
  {
    int total = E * 8;
    k_softmax_num<<<(total + 255) / 256, 256, 0, stream>>>(ebuf, mbuf, sbuf,
                                                           dst, E);
  }
  // 4) normalize + scatter-sum messages
  {
    long long total = (long long)E * 64;
    int blocks = (int)((total + 255) / 256);
    k_scatter<<<blocks, 256, 0, stream>>>(ebuf, sbuf, hbuf, src, dst, h_out, E);
  }
}